// ChebNet_25134148616266
// MI455X (gfx1250) — compile-verified
//
#include <hip/hip_runtime.h>
#include <hip/hip_bf16.h>

typedef __attribute__((ext_vector_type(2))) float v2f;
typedef __attribute__((ext_vector_type(8))) float v8f;

// -------------------- elementwise / scatter kernels --------------------

__global__ void fill0_k(float* __restrict__ p, int n) {
  int i = blockIdx.x * blockDim.x + threadIdx.x;
  if (i < n) p[i] = 0.0f;
}

__global__ void degree_k(const int* __restrict__ dst, float* __restrict__ deg, int E) {
  int e = blockIdx.x * blockDim.x + threadIdx.x;
  if (e < E) atomicAdd(&deg[dst[e]], 1.0f);
}

__global__ void rsqrt_k(float* __restrict__ deg, int n) {
  int i = blockIdx.x * blockDim.x + threadIdx.x;
  if (i < n) {
    float v = deg[i];
    deg[i] = (v > 0.0f) ? rsqrtf(v) : 0.0f;
  }
}

// tx[d,:] += -(dinv[s]*dinv[d]) * x[s,:]   one thread per (edge, feature).
// F is a compile-time power of two -> idx/F is a shift, idx%F a mask.
template <int F>
__global__ void scatter_k(const float* __restrict__ x, const int* __restrict__ src,
                          const int* __restrict__ dst, const float* __restrict__ dinv,
                          float* __restrict__ tx, int total) {
  int idx = blockIdx.x * blockDim.x + threadIdx.x;
  if (idx >= total) return;
  int e = idx / F;
  int f = idx & (F - 1);
  int s = src[e], d = dst[e];
  float w = -(dinv[s] * dinv[d]);
  atomicAdd(&tx[d * F + f], w * x[s * F + f]);
}

// segment_max of non-negative floats: integer atomicMax on the bit pattern
// (valid because inputs are post-ReLU >= 0 and dest is zero-initialized)
template <int F>
__global__ void segmax_k(const float* __restrict__ h, const int* __restrict__ cluster,
                         float* __restrict__ out, int total) {
  int idx = blockIdx.x * blockDim.x + threadIdx.x;
  if (idx >= total) return;
  int i = idx / F;
  int f = idx & (F - 1);
  atomicMax((unsigned int*)&out[cluster[i] * F + f], __float_as_uint(h[idx]));
}

template <int F>
__global__ void batchsum_k(const float* __restrict__ h3, const int* __restrict__ batch,
                           float* __restrict__ sums, float* __restrict__ cnt, int total) {
  int idx = blockIdx.x * blockDim.x + threadIdx.x;
  if (idx >= total) return;
  int i = idx / F;
  int f = idx & (F - 1);
  int b = batch[i];
  atomicAdd(&sums[b * F + f], h3[idx]);
  if (f == 0) atomicAdd(&cnt[b], 1.0f);
}

// gg[g,j] = relu( (sums[g,:]/cnt[g]) @ W[:,j] + bias[j] )   (tiny: G*H1 outputs)
__global__ void fc1_k(const float* __restrict__ sums, const float* __restrict__ cnt,
                      const float* __restrict__ W, const float* __restrict__ bias,
                      float* __restrict__ gg, int G, int K, int H) {
  int idx = blockIdx.x * blockDim.x + threadIdx.x;
  if (idx >= G * H) return;
  int g = idx / H;
  int j = idx - g * H;
  float inv = 1.0f / cnt[g];
  float acc = bias[j];
  for (int k = 0; k < K; ++k) acc += sums[g * K + k] * inv * W[k * H + j];
  gg[idx] = acc > 0.0f ? acc : 0.0f;
}

// -------------------- WMMA GEMM kernels (f32 16x16x4) --------------------
// ChebConv K=2 fused: H = relu(X@W0 + TX@W1 + bias).
// 4 waves/block, one 16x16 output tile per wave. TX and H live in workspace
// padded (rows rounded up to 64) and zero-filled, so their accesses are
// unconditional. X is clamp-loaded + cndmask'ed when XSAFE=false.
// K and F are compile-time: all W/H offsets become immediates.
template <int K, int F, bool XSAFE>
__global__ void cheb_wmma_k(const float* __restrict__ X, const float* __restrict__ TX,
                            const float* __restrict__ W0, const float* __restrict__ W1,
                            const float* __restrict__ bias, float* __restrict__ H,
                            int n) {
  const int wid  = threadIdx.x >> 5;
  const int lane = threadIdx.x & 31;
  const int row0 = (blockIdx.x * 4 + wid) * 16;
  const int c0   = blockIdx.y * 16;
  const int half = lane >> 4;      // 0: K-pair {0,1}, 1: K-pair {2,3}
  const int lm   = lane & 15;
  const int row  = row0 + lm;      // A: M in lanes
  const int col  = c0 + lm;        // B: N in lanes
  const bool rv  = row < n;
  const int rowc = XSAFE ? row : (rv ? row : n - 1);

  const float* __restrict__ xr  = X + rowc * K;
  const float* __restrict__ tr  = TX + row * K;
  const float* __restrict__ w0c = W0 + col;        // + ka*F immediate
  const float* __restrict__ w1c = W1 + col;

  v8f acc = {};
#pragma unroll
  for (int k0 = 0; k0 < K; k0 += 4) {
    const int ka = k0 + half * 2;
    v2f a = *(const v2f*)(xr + ka);
    if (!XSAFE) { a.x = rv ? a.x : 0.0f; a.y = rv ? a.y : 0.0f; }
    v2f b;
    b.x = w0c[ka * F];
    b.y = w0c[(ka + 1) * F];
    acc = __builtin_amdgcn_wmma_f32_16x16x4_f32(false, a, false, b, (short)0, acc, false, false);
  }
#pragma unroll
  for (int k0 = 0; k0 < K; k0 += 4) {
    const int ka = k0 + half * 2;
    v2f a = *(const v2f*)(tr + ka);   // padded+zeroed workspace: always valid
    v2f b;
    b.x = w1c[ka * F];
    b.y = w1c[(ka + 1) * F];
    acc = __builtin_amdgcn_wmma_f32_16x16x4_f32(false, a, false, b, (short)0, acc, false, false);
  }
  const float bb = bias[col];
  float* __restrict__ hb = H + (row0 + half * 8) * F + col;  // + r*F immediate
#pragma unroll
  for (int r = 0; r < 8; ++r) {
    float v = acc[r] + bb;                 // C/D: VGPR r -> M = r + 8*laneHalf
    hb[r * F] = v > 0.0f ? v : 0.0f;       // H padded: unconditional store
  }
}

// out[m, col] = gg[m,:] @ W[:,col] + bias[col]; gg padded/zeroed to 16 rows.
template <int K>
__global__ void fc2_wmma_k(const float* __restrict__ gg, const float* __restrict__ W,
                           const float* __restrict__ bias, float* __restrict__ out,
                           int G, int OUT) {
  const int wid  = threadIdx.x >> 5;
  const int lane = threadIdx.x & 31;
  const int c0   = (blockIdx.x * 4 + wid) * 16;
  const int half = lane >> 4;
  const int lm   = lane & 15;
  const int col  = c0 + lm;
  const int colc = col < OUT ? col : OUT - 1;   // clamp: load always valid
  const float* __restrict__ gr = gg + lm * K;
  const float* __restrict__ wc = W + colc;

  v8f acc = {};
#pragma unroll
  for (int k0 = 0; k0 < K; k0 += 4) {
    const int ka = k0 + half * 2;
    v2f a = *(const v2f*)(gr + ka);
    v2f b;
    b.x = wc[ka * OUT];
    b.y = wc[(ka + 1) * OUT];
    acc = __builtin_amdgcn_wmma_f32_16x16x4_f32(false, a, false, b, (short)0, acc, false, false);
  }
  const float bb = bias[colc];
  if (col < OUT) {
#pragma unroll
    for (int r = 0; r < 8; ++r) {
      int m = r + half * 8;
      if (m < G) out[m * OUT + col] = acc[r] + bb;
    }
  }
}

// -------------------- launch --------------------

static inline size_t r64(size_t n) { return (n + 63) & ~(size_t)63; }

extern "C" void kernel_launch(void* const* d_in, const int* in_sizes, int n_in,
                              void* d_out, int out_size, void* d_ws, size_t ws_size,
                              hipStream_t stream) {
  const float* x    = (const float*)d_in[0];
  const float* W0a  = (const float*)d_in[1];
  const float* W1a  = (const float*)d_in[2];
  const float* b1   = (const float*)d_in[3];
  const float* W0b  = (const float*)d_in[4];
  const float* W1b  = (const float*)d_in[5];
  const float* b2   = (const float*)d_in[6];
  const float* fcw1 = (const float*)d_in[7];
  const float* fcb1 = (const float*)d_in[8];
  const float* fcw2 = (const float*)d_in[9];
  const float* fcb2 = (const float*)d_in[10];
  const int* ei1      = (const int*)d_in[11];
  const int* cluster1 = (const int*)d_in[12];
  const int* ei2      = (const int*)d_in[13];
  const int* cluster2 = (const int*)d_in[14];
  const int* batch3   = (const int*)d_in[15];

  const int N   = in_sizes[12];          // cluster1 length = 80000
  const int E1  = in_sizes[11] / 2;
  const int n1  = in_sizes[14];          // cluster2 length
  const int E2  = in_sizes[13] / 2;
  const int n2  = in_sizes[15];          // batch3 length
  const int C1  = in_sizes[3];           // 32
  const int C2  = in_sizes[6];           // 64
  const int H1  = in_sizes[8];           // 128
  const int OUT = in_sizes[10];          // 20000
  const int G   = out_size / OUT;        // 8
  const int FIN = in_sizes[1] / C1;      // 16

  // carve workspace (256B aligned, node dims padded to 64 rows for guard-free WMMA I/O)
  char* p = (char*)d_ws;
  auto carve = [&](size_t nf) {
    float* r = (float*)p;
    p += ((nf * sizeof(float)) + 255) & ~(size_t)255;
    return r;
  };
  const size_t Np  = r64((size_t)N);
  const size_t n1p = r64((size_t)n1);
  float* deg1 = carve((size_t)N);
  float* tx1  = carve(Np * FIN);
  float* h1   = carve(Np * C1);
  float* hp   = carve(n1p * C1);
  float* deg2 = carve((size_t)n1);
  float* tx2  = carve(n1p * C1);
  float* h2   = carve(n1p * C2);
  float* h3   = carve((size_t)n2 * C2);
  float* sums = carve((size_t)G * C2);
  float* cnt  = carve((size_t)G);
  float* gg   = carve((size_t)16 * H1);   // padded to 16 rows for fc2 A-loads

  const int totalF = (int)(((char*)p - (char*)d_ws) / sizeof(float));
  const int B = 256;

  // 0) zero all scratch (atomics accumulate into it; padding must be zero)
  fill0_k<<<(totalF + B - 1) / B, B, 0, stream>>>((float*)d_ws, totalF);

  // ---- layer 1 ----
  degree_k<<<(E1 + B - 1) / B, B, 0, stream>>>(ei1 + E1, deg1, E1);
  rsqrt_k<<<(N + B - 1) / B, B, 0, stream>>>(deg1, N);
  scatter_k<16><<<((E1 * 16) + B - 1) / B, B, 0, stream>>>(x, ei1, ei1 + E1, deg1, tx1,
                                                           E1 * 16);
  dim3 g1(((N + 15) / 16 + 3) / 4, 2);
  cheb_wmma_k<16, 32, false><<<g1, 128, 0, stream>>>(x, tx1, W0a, W1a, b1, h1, N);
  segmax_k<32><<<((N * 32) + B - 1) / B, B, 0, stream>>>(h1, cluster1, hp, N * 32);

  // ---- layer 2 ----
  degree_k<<<(E2 + B - 1) / B, B, 0, stream>>>(ei2 + E2, deg2, E2);
  rsqrt_k<<<(n1 + B - 1) / B, B, 0, stream>>>(deg2, n1);
  scatter_k<32><<<((E2 * 32) + B - 1) / B, B, 0, stream>>>(hp, ei2, ei2 + E2, deg2, tx2,
                                                           E2 * 32);
  dim3 g2(((n1 + 15) / 16 + 3) / 4, 4);
  cheb_wmma_k<32, 64, true><<<g2, 128, 0, stream>>>(hp, tx2, W0b, W1b, b2, h2, n1);
  segmax_k<64><<<((n1 * 64) + B - 1) / B, B, 0, stream>>>(h2, cluster2, h3, n1 * 64);

  // ---- readout ----
  batchsum_k<64><<<((n2 * 64) + B - 1) / B, B, 0, stream>>>(h3, batch3, sums, cnt, n2 * 64);
  fc1_k<<<((G * H1) + B - 1) / B, B, 0, stream>>>(sums, cnt, fcw1, fcb1, gg, G, C2, H1);
  fc2_wmma_k<128><<<((OUT + 15) / 16 + 3) / 4, 128, 0, stream>>>(gg, fcw2, fcb2,
                                                                 (float*)d_out, G, OUT);
}